// CGNN_GAT_70214125355156
// MI455X (gfx1250) — compile-verified
//
#include <hip/hip_runtime.h>
#include <hip/hip_fp16.h>
#include <math.h>

typedef __attribute__((ext_vector_type(2))) float v2f;
typedef __attribute__((ext_vector_type(8))) float v8f;

#define N_NODES 50000
#define E_EDGES 800000
#define ETOT    (E_EDGES + N_NODES)
#define F_IN    128
#define H_GCN   512
#define F_GAT   256   // HEADS * F_H
#define F_H     64

// ---------------- atomics helpers ----------------
__device__ __forceinline__ void atomicAddF32(float* p, float v) {
  unsafeAtomicAdd(p, v);   // -> global_atomic_add_f32 (near-memory L2 atomic)
}
__device__ __forceinline__ void atomicMaxF32(float* p, float v) {
  if (v >= 0.0f) atomicMax((int*)p, __float_as_int(v));
  else           atomicMin((unsigned int*)p, __float_as_uint(v));
}
__device__ __forceinline__ float lrelu(float x, float s) {
  return x >= 0.0f ? x : s * x;
}

// ---------------- fp32 WMMA GEMM: C[M,NC] = actA(A[M,K]) @ B[K,NC] ----------
// one wave per 16x16 output tile; K stepped by 4 with v_wmma_f32_16x16x4_f32
// A frag (16x4 f32): lane M = lane&15, K = 2*(lane>>4) + {0,1}  -> v2f
// B frag (4x16 f32): lane N = lane&15, K = 2*(lane>>4) + {0,1}  -> v2f
template<int K, int NC, bool FUSE_BIAS_LEAKY>
__global__ __launch_bounds__(256) void gemm_wmma(const float* __restrict__ A,
                                                 const float* __restrict__ B,
                                                 const float* __restrict__ bias,
                                                 float* __restrict__ C,
                                                 int tiles_m, int tiles_n) {
  const int wave = threadIdx.x >> 5;
  const int lane = threadIdx.x & 31;
  const int tile = blockIdx.x * 8 + wave;
  if (tile >= tiles_m * tiles_n) return;
  const int tm   = tile / tiles_n;
  const int tn   = tile % tiles_n;
  const int half = lane >> 4;      // 0: lanes 0-15, 1: lanes 16-31
  const int l15  = lane & 15;
  const int row  = tm * 16 + l15;  // A row held by this lane
  const int col  = tn * 16 + l15;  // B/C column held by this lane

  v8f c = {};
#pragma unroll 4
  for (int k0 = 0; k0 < K; k0 += 4) {
    const int ka = k0 + half * 2;
    v2f a, b;
    float x0 = A[row * K + ka];
    float x1 = A[row * K + ka + 1];
    if constexpr (FUSE_BIAS_LEAKY) {
      x0 = lrelu(x0 + bias[ka],     0.01f);
      x1 = lrelu(x1 + bias[ka + 1], 0.01f);
    }
    a[0] = x0; a[1] = x1;
    b[0] = B[ka * NC + col];
    b[1] = B[(ka + 1) * NC + col];
    c = __builtin_amdgcn_wmma_f32_16x16x4_f32(false, a, false, b,
                                              (short)0, c, false, false);
  }
  // C/D layout: VGPR r -> M = r + 8*half, N = lane&15
#pragma unroll
  for (int r = 0; r < 8; ++r)
    C[(tm * 16 + r + half * 8) * NC + col] = c[r];
}

// ---------------- degree / normalization ----------------
__global__ __launch_bounds__(256) void deg_init(float* deg) {
  int i = blockIdx.x * 256 + threadIdx.x;
  if (i < N_NODES) deg[i] = 1.0f;                 // self-loop counted here
}
__global__ __launch_bounds__(256) void deg_edges(const int* __restrict__ dst,
                                                 float* deg) {
  int e = blockIdx.x * 256 + threadIdx.x;
  if (e < E_EDGES) atomicAddF32(&deg[dst[e]], 1.0f);
}
__global__ __launch_bounds__(256) void deg_to_dinv(float* deg) {
  int i = blockIdx.x * 256 + threadIdx.x;
  if (i < N_NODES) deg[i] = rsqrtf(deg[i]);
}
// h1 init = self-loop contribution dinv(n)^2 * xw(n)
__global__ __launch_bounds__(256) void h1_init(const float* __restrict__ xw,
                                               const float* __restrict__ dinv,
                                               float* __restrict__ h1) {
  int idx = blockIdx.x * 256 + threadIdx.x;     // N*512 threads
  int n = idx >> 9;
  float dn = dinv[n];
  h1[idx] = dn * dn * xw[idx];
}
// GCN edge scatter: one wave per edge, 16 coalesced channels per lane
__global__ __launch_bounds__(256) void gcn_scatter(const int* __restrict__ src,
                                                   const int* __restrict__ dst,
                                                   const float* __restrict__ dinv,
                                                   const float* __restrict__ xw,
                                                   float* __restrict__ h1) {
  int e = blockIdx.x * 8 + (threadIdx.x >> 5);
  if (e >= E_EDGES) return;
  int lane = threadIdx.x & 31;
  int s = src[e], d = dst[e];
  float nrm = dinv[s] * dinv[d];
  const float* xr = xw + (size_t)s * H_GCN;
  float*       hr = h1 + (size_t)d * H_GCN;
#pragma unroll
  for (int i = 0; i < 16; ++i) {
    int c = i * 32 + lane;
    atomicAddF32(&hr[c], nrm * xr[c]);
  }
}

// ---------------- GAT attention ----------------
__global__ __launch_bounds__(256) void att_scalars(const float* __restrict__ h2,
                                                   const float* __restrict__ att_s,
                                                   const float* __restrict__ att_d,
                                                   float* __restrict__ a_src,
                                                   float* __restrict__ a_dst) {
  int idx = blockIdx.x * 256 + threadIdx.x;     // N*4
  if (idx >= N_NODES * 4) return;
  int n = idx >> 2, h = idx & 3;
  const float* hr = h2 + (size_t)n * F_GAT + h * F_H;
  float s = 0.f, d = 0.f;
#pragma unroll 8
  for (int f = 0; f < F_H; ++f) {
    float v = hr[f];
    s += v * att_s[h * F_H + f];
    d += v * att_d[h * F_H + f];
  }
  a_src[idx] = s; a_dst[idx] = d;
}
__global__ __launch_bounds__(256) void init_att(float* __restrict__ oa,
                                                float* __restrict__ e_max,
                                                float* __restrict__ denom) {
  int idx = blockIdx.x * 256 + threadIdx.x;     // N*256
  oa[idx] = 0.0f;
  if (idx < N_NODES * 4) {
    ((unsigned int*)e_max)[idx] = 0xFF800000u;  // -inf
    denom[idx] = 0.0f;
  }
}
__device__ __forceinline__ void edge_sd(int idx, const int* src, const int* dst,
                                        int& s, int& d) {
  if (idx < E_EDGES) { s = src[idx]; d = dst[idx]; }
  else               { s = d = idx - E_EDGES; }   // appended self-loop
}
__global__ __launch_bounds__(256) void att_edge_max(const int* __restrict__ src,
                                                    const int* __restrict__ dst,
                                                    const float* __restrict__ a_src,
                                                    const float* __restrict__ a_dst,
                                                    float* __restrict__ e_max) {
  int idx = blockIdx.x * 256 + threadIdx.x;     // ETOT*4
  if (idx >= ETOT * 4) return;
  int e = idx >> 2, h = idx & 3, s, d;
  edge_sd(e, src, dst, s, d);
  float v = lrelu(a_src[s * 4 + h] + a_dst[d * 4 + h], 0.2f);
  atomicMaxF32(&e_max[d * 4 + h], v);
}
__global__ __launch_bounds__(256) void att_edge_denom(const int* __restrict__ src,
                                                      const int* __restrict__ dst,
                                                      const float* __restrict__ a_src,
                                                      const float* __restrict__ a_dst,
                                                      const float* __restrict__ e_max,
                                                      float* __restrict__ denom) {
  int idx = blockIdx.x * 256 + threadIdx.x;     // ETOT*4
  if (idx >= ETOT * 4) return;
  int e = idx >> 2, h = idx & 3, s, d;
  edge_sd(e, src, dst, s, d);
  float v = lrelu(a_src[s * 4 + h] + a_dst[d * 4 + h], 0.2f);
  atomicAddF32(&denom[d * 4 + h], expf(v - e_max[d * 4 + h]));
}
// weighted aggregate: one wave per edge, 8 coalesced channels per lane
__global__ __launch_bounds__(256) void att_edge_agg(const int* __restrict__ src,
                                                    const int* __restrict__ dst,
                                                    const float* __restrict__ a_src,
                                                    const float* __restrict__ a_dst,
                                                    const float* __restrict__ e_max,
                                                    const float* __restrict__ denom,
                                                    const float* __restrict__ h2,
                                                    float* __restrict__ oa) {
  int e = blockIdx.x * 8 + (threadIdx.x >> 5);
  if (e >= ETOT) return;
  int lane = threadIdx.x & 31, s, d;
  edge_sd(e, src, dst, s, d);
  float alpha[4];
#pragma unroll
  for (int h = 0; h < 4; ++h) {
    float v = lrelu(a_src[s * 4 + h] + a_dst[d * 4 + h], 0.2f);
    alpha[h] = expf(v - e_max[d * 4 + h]) / (denom[d * 4 + h] + 1e-16f);
  }
  const float* hr = h2 + (size_t)s * F_GAT;
  float*       od = oa + (size_t)d * F_GAT;
#pragma unroll
  for (int i = 0; i < 8; ++i) {
    int c = i * 32 + lane;
    atomicAddF32(&od[c], alpha[c >> 6] * hr[c]);
  }
}
// head-mean + bias + leaky + FC, one wave per node with wave32 reduction
__global__ __launch_bounds__(256) void finalize_k(const float* __restrict__ oa,
                                                  const float* __restrict__ bg,
                                                  const float* __restrict__ wfc,
                                                  const float* __restrict__ bfc,
                                                  float* __restrict__ out) {
  int n = blockIdx.x * 8 + (threadIdx.x >> 5);
  if (n >= N_NODES) return;
  int lane = threadIdx.x & 31;
  const float* r = oa + (size_t)n * F_GAT;
  float acc = 0.f;
#pragma unroll
  for (int j = 0; j < 2; ++j) {
    int f = j * 32 + lane;
    float m = 0.25f * (r[f] + r[64 + f] + r[128 + f] + r[192 + f]) + bg[f];
    acc += lrelu(m, 0.01f) * wfc[f];
  }
#pragma unroll
  for (int off = 16; off > 0; off >>= 1) acc += __shfl_down(acc, off, 32);
  if (lane == 0) out[n] = acc + bfc[0];
}

// ---------------- launcher ----------------
extern "C" void kernel_launch(void* const* d_in, const int* in_sizes, int n_in,
                              void* d_out, int out_size, void* d_ws, size_t ws_size,
                              hipStream_t stream) {
  const float* x    = (const float*)d_in[0];
  const int*   ei   = (const int*)  d_in[1];
  const float* w1   = (const float*)d_in[2];
  const float* b1   = (const float*)d_in[3];
  const float* wg   = (const float*)d_in[4];
  const float* atts = (const float*)d_in[5];
  const float* attd = (const float*)d_in[6];
  const float* bg   = (const float*)d_in[7];
  const float* wfc  = (const float*)d_in[8];
  const float* bfc  = (const float*)d_in[9];
  const int* src = ei;
  const int* dst = ei + E_EDGES;

  float* bufA  = (float*)d_ws;                       // xw -> later h2|oa
  float* bufB  = bufA + (size_t)N_NODES * H_GCN;     // h1
  float* deg   = bufB + (size_t)N_NODES * H_GCN;     // deg -> dinv (in place)
  float* a_src = deg   + N_NODES;
  float* a_dst = a_src + N_NODES * 4;
  float* e_max = a_dst + N_NODES * 4;
  float* denom = e_max + N_NODES * 4;
  float* h2 = bufA;
  float* oa = bufA + (size_t)N_NODES * F_GAT;

  dim3 b(256);
  // GCN: xw = x @ w1   (3125 x 32 tiles, 8 waves/block)
  gemm_wmma<F_IN, H_GCN, false><<<(3125 * 32) / 8, b, 0, stream>>>(
      x, w1, nullptr, bufA, 3125, 32);
  deg_init   <<<(N_NODES + 255) / 256, b, 0, stream>>>(deg);
  deg_edges  <<<(E_EDGES + 255) / 256, b, 0, stream>>>(dst, deg);
  deg_to_dinv<<<(N_NODES + 255) / 256, b, 0, stream>>>(deg);
  h1_init    <<<(N_NODES * H_GCN) / 256, b, 0, stream>>>(bufA, deg, bufB);
  gcn_scatter<<<(E_EDGES + 7) / 8, b, 0, stream>>>(src, dst, deg, bufA, bufB);
  // GAT: h2 = leaky(h1 + b1) @ wg   (bias+act fused into A load)
  gemm_wmma<H_GCN, F_GAT, true><<<(3125 * 16) / 8, b, 0, stream>>>(
      bufB, wg, b1, h2, 3125, 16);
  att_scalars<<<(N_NODES * 4 + 255) / 256, b, 0, stream>>>(h2, atts, attd, a_src, a_dst);
  init_att   <<<(N_NODES * F_GAT) / 256, b, 0, stream>>>(oa, e_max, denom);
  att_edge_max  <<<(ETOT * 4 + 255) / 256, b, 0, stream>>>(src, dst, a_src, a_dst, e_max);
  att_edge_denom<<<(ETOT * 4 + 255) / 256, b, 0, stream>>>(src, dst, a_src, a_dst, e_max, denom);
  att_edge_agg  <<<(ETOT + 7) / 8, b, 0, stream>>>(src, dst, a_src, a_dst, e_max, denom, h2, oa);
  finalize_k    <<<(N_NODES + 7) / 8, b, 0, stream>>>(oa, bg, wfc, bfc, (float*)d_out);
}